// MultiLevelHashEncoding_55353538511352
// MI455X (gfx1250) — compile-verified
//
#include <hip/hip_runtime.h>
#include <stdint.h>

// ---------------------------------------------------------------------------
// MultiLevelHashEncoding for MI455X (gfx1250)
//   - pre-pass interleaves every level's (2, n_enc) table into float2 pairs
//     in d_ws (halves the gather count of the memory-bound main kernel)
//   - level-0 table (32 KB) staged into LDS via the CDNA5 Tensor Data Mover
//     (tensor_load_to_lds + s_wait_tensorcnt), gathered with ds_load_b64
//   - one point per thread, level 0 peeled (pure LDS), levels 1-15 unrolled
// ---------------------------------------------------------------------------

#define N_LEVELS 16
#define NPTS     (1u << 20)
#define HASH_MASK 0x7FFFFu
#define P1 2654435761u   // y-dim prime
#define P2 805459861u    // x-dim prime

static constexpr int  kRes[N_LEVELS]   = {16,20,25,32,40,50,64,80,101,128,161,203,256,322,406,512};
static constexpr int  kNenc[N_LEVELS]  = {4096,8000,15625,32768,64000,125000,262144,512000,
                                          524288,524288,524288,524288,524288,524288,524288,524288};
static constexpr bool kDense[N_LEVELS] = {true,true,true,true,true,true,true,true,
                                          false,false,false,false,false,false,false,false};
static constexpr int  kOff[N_LEVELS]   = {0,4096,12096,27721,60489,124489,249489,511633,
                                          1023633,1547921,2072209,2596497,3120785,3645073,4169361,4693649};
static constexpr int  kTotal           = 5217937;   // sum of kNenc (float2 entries)

struct EmbPtrs { const float* p[N_LEVELS]; };

typedef unsigned int uint32x4 __attribute__((ext_vector_type(4)));
typedef int          int32x8  __attribute__((ext_vector_type(8)));
typedef int          int32x4  __attribute__((ext_vector_type(4)));

// ---------------------------------------------------------------------------
// Pre-pass: repack emb[l] from planar (2, n_enc) to interleaved float2.
// ---------------------------------------------------------------------------
__global__ __launch_bounds__(256)
void hashenc_interleave(EmbPtrs ep, float2* __restrict__ tab) {
    int stride = gridDim.x * blockDim.x;
    for (int e = blockIdx.x * blockDim.x + threadIdx.x; e < kTotal; e += stride) {
        int l = 0;
        #pragma unroll
        for (int i = 1; i < N_LEVELS; ++i)
            if (e >= kOff[i]) l = i;
        int idx = e - kOff[l];
        const float* src = ep.p[l];
        tab[e] = make_float2(src[idx], src[idx + kNenc[l]]);
    }
}

// ---------------------------------------------------------------------------
// Per-dimension corner setup: two weights (validity folded in) + two indices.
// pos in [-0.5, res-0.5]; corner0 = floor(pos) in [-1, res-1]; corner1 in [0, res].
// ---------------------------------------------------------------------------
__device__ __forceinline__
void corner_setup(float x, float resf, float& wa, float& wb, int& ia, int& ib) {
    float pos = 0.5f * ((x + 1.0f) * resf - 1.0f);
    float p0  = floorf(pos);
    float f   = pos - p0;
    wa = 1.0f - f;
    wb = f;
    if (p0 < 0.0f)          wa = 0.0f;   // corner0 == -1  -> zero pad
    if (p0 > resf - 2.0f)   wb = 0.0f;   // corner1 == res -> zero pad
    ia = (int)fmaxf(p0, 0.0f);
    ib = (int)fminf(p0 + 1.0f, resf - 1.0f);
}

// Per-level corner index/weight bundle.
struct Corners {
    unsigned px[2], py[2], pz[2];
    float    wX[2], wY[2], wZ[2];
};

template <int RES, bool DENSE>
__device__ __forceinline__
Corners make_corners(float x0, float x1, float x2) {
    Corners c;
    const float resf = (float)RES;
    float wxa, wxb, wya, wyb, wza, wzb;
    int   ixa, ixb, iya, iyb, iza, izb;
    corner_setup(x0, resf, wxa, wxb, ixa, ixb);
    corner_setup(x1, resf, wya, wyb, iya, iyb);
    corner_setup(x2, resf, wza, wzb, iza, izb);
    if (DENSE) {
        c.px[0] = (unsigned)ixa;                 c.px[1] = (unsigned)ixb;
        c.py[0] = (unsigned)iya * (unsigned)RES; c.py[1] = (unsigned)iyb * (unsigned)RES;
        c.pz[0] = (unsigned)iza * (unsigned)(RES * RES);
        c.pz[1] = (unsigned)izb * (unsigned)(RES * RES);
    } else {
        c.px[0] = (unsigned)ixa * P2;  c.px[1] = (unsigned)ixb * P2;
        c.py[0] = (unsigned)iya * P1;  c.py[1] = (unsigned)iyb * P1;
        c.pz[0] = (unsigned)iza;       c.pz[1] = (unsigned)izb;
    }
    c.wX[0] = wxa; c.wX[1] = wxb;
    c.wY[0] = wya; c.wY[1] = wyb;
    c.wZ[0] = wza; c.wZ[1] = wzb;
    return c;
}

__device__ __forceinline__
unsigned corner_index(const Corners& c, int cx, int cy, int cz, bool dense) {
    return dense ? (c.pz[cz] + c.py[cy] + c.px[cx])
                 : ((c.pz[cz] ^ c.py[cy] ^ c.px[cx]) & HASH_MASK);
}

// ---------------------------------------------------------------------------
// Main kernel (interleaved path): level-0 table staged in LDS via TDM.
// ---------------------------------------------------------------------------
__global__ __launch_bounds__(256)
void hashenc_main(const float* __restrict__ x, const float2* __restrict__ tab,
                  float2* __restrict__ out) {
    __shared__ float2 lds0[4096];      // 32 KB, first (only) LDS object -> offset 0

    // ---- Tensor Data Mover: DMA 4096 x 8B from tab (level 0) into LDS ----
    if ((threadIdx.x >> 5) == 0) {     // one wave issues the descriptor
        uint64_t ga = (uint64_t)(uintptr_t)tab;       // kOff[0] == 0
        uint32x4 g0;
        g0.x = 1u;                                    // count=1 valid descriptor
        g0.y = 0u;                                    // lds_addr = 0 (lds0 base)
        g0.z = (unsigned)(ga & 0xFFFFFFFFu);          // global_addr[31:0]
        g0.w = (unsigned)((ga >> 32) & 0x1FFFFFFu)    // global_addr[56:32]
             | (2u << 30);                            // type = 2 ("image")
        int32x8 g1;
        g1[0] = (int)(3u << 16);       // data_size = 3 (8 bytes/elem)
        g1[1] = (int)(4096u << 16);    // tensor_dim0 low16 (bits 63:48)
        g1[2] = (int)(1u << 16);       // tensor_dim0 hi=0 | tensor_dim1 low16 = 1
        g1[3] = (int)(4096u << 16);    // tensor_dim1 hi=0 | tile_dim0 = 4096
        g1[4] = 1;                     // tile_dim1 = 1, tile_dim2 = 0
        g1[5] = 4096;                  // tensor_dim0_stride low32
        g1[6] = 0;                     // stride0 hi | stride1 low
        g1[7] = 0;
        int32x4 gz4 = {0, 0, 0, 0};
        int32x8 gz8 = {0, 0, 0, 0, 0, 0, 0, 0};
        __builtin_amdgcn_tensor_load_to_lds(g0, g1, gz4, gz4, gz8, 0);
        __builtin_amdgcn_s_wait_tensorcnt(0);
    }
    __syncthreads();
    // The TDM write to LDS is invisible to alias analysis (lds_addr is an
    // integer inside the descriptor). Escape lds0's address into an asm with
    // a memory clobber so the compiler must assume the array was written and
    // cannot fold its loads to undef.
    {
        float2* escape = lds0;
        asm volatile("" : "+v"(escape) :: "memory");
    }

    unsigned b = blockIdx.x * 256u + threadIdx.x;     // grid covers NPTS exactly
    float x0 = x[3u * b + 0u];
    float x1 = x[3u * b + 1u];
    float x2 = x[3u * b + 2u];
    float2* outp = out + (size_t)b * N_LEVELS;

    // ---- Level 0: dedicated LDS gathers (ds_load_b64) -------------------
    {
        Corners c = make_corners<16, true>(x0, x1, x2);
        float ax = 0.0f, ay = 0.0f;
        #pragma unroll
        for (int k = 0; k < 8; ++k) {
            const int cx = k & 1, cy = (k >> 1) & 1, cz = (k >> 2) & 1;
            float w = c.wX[cx] * c.wY[cy] * c.wZ[cz];
            unsigned idx = corner_index(c, cx, cy, cz, true);
            float2 g = lds0[idx];                       // addrspace(3) load
            ax = fmaf(w, g.x, ax);
            ay = fmaf(w, g.y, ay);
        }
        outp[0] = make_float2(ax, ay);
    }

    // ---- Levels 1..15: interleaved global gathers -----------------------
    #pragma unroll
    for (int l = 1; l < N_LEVELS; ++l) {
        Corners c;
        if (kDense[l]) {
            switch (l) {   // constant RES template instantiation per level
                case 1: c = make_corners<20,  true>(x0, x1, x2); break;
                case 2: c = make_corners<25,  true>(x0, x1, x2); break;
                case 3: c = make_corners<32,  true>(x0, x1, x2); break;
                case 4: c = make_corners<40,  true>(x0, x1, x2); break;
                case 5: c = make_corners<50,  true>(x0, x1, x2); break;
                case 6: c = make_corners<64,  true>(x0, x1, x2); break;
                default: c = make_corners<80, true>(x0, x1, x2); break;
            }
        } else {
            switch (l) {
                case 8:  c = make_corners<101, false>(x0, x1, x2); break;
                case 9:  c = make_corners<128, false>(x0, x1, x2); break;
                case 10: c = make_corners<161, false>(x0, x1, x2); break;
                case 11: c = make_corners<203, false>(x0, x1, x2); break;
                case 12: c = make_corners<256, false>(x0, x1, x2); break;
                case 13: c = make_corners<322, false>(x0, x1, x2); break;
                case 14: c = make_corners<406, false>(x0, x1, x2); break;
                default: c = make_corners<512, false>(x0, x1, x2); break;
            }
        }
        const float2* lt = tab + (unsigned)kOff[l];
        float ax = 0.0f, ay = 0.0f;
        #pragma unroll
        for (int k = 0; k < 8; ++k) {
            const int cx = k & 1, cy = (k >> 1) & 1, cz = (k >> 2) & 1;
            float w = c.wX[cx] * c.wY[cy] * c.wZ[cz];
            unsigned idx = corner_index(c, cx, cy, cz, kDense[l]);
            float2 g = lt[idx];
            ax = fmaf(w, g.x, ax);
            ay = fmaf(w, g.y, ay);
        }
        outp[l] = make_float2(ax, ay);
    }
}

// ---------------------------------------------------------------------------
// Fallback (ws too small): gather the two planes directly, no LDS staging.
// ---------------------------------------------------------------------------
__global__ __launch_bounds__(256)
void hashenc_direct(const float* __restrict__ x, EmbPtrs ep, float2* __restrict__ out) {
    unsigned b = blockIdx.x * 256u + threadIdx.x;
    float x0 = x[3u * b + 0u];
    float x1 = x[3u * b + 1u];
    float x2 = x[3u * b + 2u];
    float2* outp = out + (size_t)b * N_LEVELS;

    #pragma unroll
    for (int l = 0; l < N_LEVELS; ++l) {
        const int   res  = kRes[l];
        const float resf = (float)res;
        const float* emb = ep.p[l];

        float wxa, wxb, wya, wyb, wza, wzb;
        int   ixa, ixb, iya, iyb, iza, izb;
        corner_setup(x0, resf, wxa, wxb, ixa, ixb);
        corner_setup(x1, resf, wya, wyb, iya, iyb);
        corner_setup(x2, resf, wza, wzb, iza, izb);

        unsigned px[2], py[2], pz[2];
        if (kDense[l]) {
            px[0] = (unsigned)ixa;                 px[1] = (unsigned)ixb;
            py[0] = (unsigned)iya * (unsigned)res; py[1] = (unsigned)iyb * (unsigned)res;
            pz[0] = (unsigned)iza * (unsigned)(res * res);
            pz[1] = (unsigned)izb * (unsigned)(res * res);
        } else {
            px[0] = (unsigned)ixa * P2;  px[1] = (unsigned)ixb * P2;
            py[0] = (unsigned)iya * P1;  py[1] = (unsigned)iyb * P1;
            pz[0] = (unsigned)iza;       pz[1] = (unsigned)izb;
        }
        float wX[2] = {wxa, wxb}, wY[2] = {wya, wyb}, wZ[2] = {wza, wzb};

        float ax = 0.0f, ay = 0.0f;
        #pragma unroll
        for (int c = 0; c < 8; ++c) {
            const int cx = c & 1, cy = (c >> 1) & 1, cz = (c >> 2) & 1;
            float w = wX[cx] * wY[cy] * wZ[cz];
            unsigned idx = kDense[l] ? (pz[cz] + py[cy] + px[cx])
                                     : ((pz[cz] ^ py[cy] ^ px[cx]) & HASH_MASK);
            ax = fmaf(w, emb[idx], ax);
            ay = fmaf(w, emb[idx + (unsigned)kNenc[l]], ay);
        }
        outp[l] = make_float2(ax, ay);
    }
}

// ---------------------------------------------------------------------------
extern "C" void kernel_launch(void* const* d_in, const int* in_sizes, int n_in,
                              void* d_out, int out_size, void* d_ws, size_t ws_size,
                              hipStream_t stream) {
    const float* x = (const float*)d_in[0];
    EmbPtrs ep;
    for (int l = 0; l < N_LEVELS; ++l)
        ep.p[l] = (const float*)d_in[1 + l];
    float2* out = (float2*)d_out;

    const size_t need = (size_t)kTotal * sizeof(float2);   // ~41.7 MB
    if (ws_size >= need) {
        float2* tab = (float2*)d_ws;
        hashenc_interleave<<<2048, 256, 0, stream>>>(ep, tab);
        hashenc_main<<<NPTS / 256, 256, 0, stream>>>(x, tab, out);
    } else {
        hashenc_direct<<<NPTS / 256, 256, 0, stream>>>(x, ep, out);
    }
}